// myLoss_45638322487920
// MI455X (gfx1250) — compile-verified
//
#include <hip/hip_runtime.h>
#include <hip/hip_bf16.h>
#include <math.h>

#define TPB 256
#define MAXN 64

typedef __attribute__((ext_vector_type(2))) float v2f;
typedef __attribute__((ext_vector_type(8))) float v8f;

__device__ __forceinline__ float iou_corner(float ax0, float ay0, float ax1, float ay1,
                                            float bx0, float by0, float bx1, float by1,
                                            float area_a, float area_b) {
  float lx = fmaxf(ax0, bx0), ly = fmaxf(ay0, by0);
  float rx = fminf(ax1, bx1), ry = fminf(ay1, by1);
  float w = fmaxf(rx - lx, 0.f), h = fmaxf(ry - ly, 0.f);
  float inter = w * h;
  return inter / (area_a + area_b - inter);
}

// ---------------------------------------------------------------------------
// Kernel 1: per-batch GT<->prior matching + smooth-L1 over positives.
// One workgroup per batch image. GT boxes cached in LDS.
// ---------------------------------------------------------------------------
__global__ void match_kernel(const float* __restrict__ priors,
                             const float* __restrict__ labels,
                             const int* __restrict__ obj_count,
                             const float* __restrict__ location,
                             int* __restrict__ conf_t,
                             int* __restrict__ num_pos,
                             float* __restrict__ accum,
                             int P, int NMAX) {
  __shared__ float gx0[MAXN], gy0[MAXN], gx1[MAXN], gy1[MAXN], gcl[MAXN], gar[MAXN];
  __shared__ unsigned long long gbest[MAXN];  // packed (iou_bits<<32 | (P-1-p))
  __shared__ int gforce[MAXN];
  __shared__ int sn;
  __shared__ float redf[TPB];
  __shared__ int   redi[TPB];

  const int b = blockIdx.x;
  const int tid = threadIdx.x;
  if (tid == 0) {
    int n = obj_count[b];
    if (n < 1) n = 1;
    if (n > NMAX) n = NMAX;
    if (n > MAXN) n = MAXN;
    sn = n;
  }
  if (tid < MAXN && tid < NMAX) {
    const float* L = labels + ((size_t)b * NMAX + tid) * 5;
    float x0 = L[0], y0 = L[1], x1 = L[2], y1 = L[3];
    gx0[tid] = x0; gy0[tid] = y0; gx1[tid] = x1; gy1[tid] = y1; gcl[tid] = L[4];
    gar[tid] = (x1 - x0) * (y1 - y0);
    gbest[tid] = (unsigned long long)(unsigned)(P - 1);  // iou=0, p=0
  }
  __syncthreads();
  const int n = sn;

  // Phase A: per-GT argmax over priors (packed u64 LDS atomicMax).
  for (int p = tid; p < P; p += TPB) {
    float cx = priors[4 * p + 0], cy = priors[4 * p + 1];
    float pw = priors[4 * p + 2], ph = priors[4 * p + 3];
    float bx0 = cx - 0.5f * pw, by0 = cy - 0.5f * ph;
    float bx1 = cx + 0.5f * pw, by1 = cy + 0.5f * ph;
    float areab = pw * ph;
    unsigned invp = (unsigned)(P - 1 - p);
    for (int g = 0; g < n; ++g) {
      float iou = iou_corner(gx0[g], gy0[g], gx1[g], gy1[g], bx0, by0, bx1, by1,
                             gar[g], areab);
      unsigned long long pk = ((unsigned long long)__float_as_uint(iou) << 32) | invp;
      if (pk > gbest[g]) atomicMax(&gbest[g], pk);
    }
  }
  __syncthreads();
  if (tid < n) gforce[tid] = P - 1 - (int)(unsigned)(gbest[tid] & 0xffffffffu);
  __syncthreads();

  // Phase B: per-prior best GT (recompute - cheaper than storing 4.5MB),
  // force-match override, conf_t, smooth-L1 over positives.
  float lloc = 0.f;
  int lpos = 0;
  for (int p = tid; p < P; p += TPB) {
    float cx = priors[4 * p + 0], cy = priors[4 * p + 1];
    float pw = priors[4 * p + 2], ph = priors[4 * p + 3];
    float bx0 = cx - 0.5f * pw, by0 = cy - 0.5f * ph;
    float bx1 = cx + 0.5f * pw, by1 = cy + 0.5f * ph;
    float areab = pw * ph;
    float bo = -1.f; int bi = 0;
    for (int g = 0; g < n; ++g) {
      float iou = iou_corner(gx0[g], gy0[g], gx1[g], gy1[g], bx0, by0, bx1, by1,
                             gar[g], areab);
      if (iou > bo) { bo = iou; bi = g; }  // first-index tie-break like argmax
    }
    for (int g = 0; g < n; ++g)
      if (gforce[g] == p) { bo = 2.f; bi = g; }  // ascending g: last wins
    int cf = (bo < 0.5f) ? 0 : ((int)gcl[bi] + 1);
    conf_t[(size_t)b * P + p] = cf;
    if (cf > 0) {
      ++lpos;
      float mx0 = gx0[bi], my0 = gy0[bi], mx1 = gx1[bi], my1 = gy1[bi];
      float tx = ((mx0 + mx1) * 0.5f - cx) / (0.1f * pw);
      float ty = ((my0 + my1) * 0.5f - cy) / (0.1f * ph);
      float tw = __logf((mx1 - mx0) / pw) * 5.0f;  // / VAR1
      float th = __logf((my1 - my0) / ph) * 5.0f;
      const float* lp = location + ((size_t)b * P + p) * 4;
      float t4[4] = {tx, ty, tw, th};
#pragma unroll
      for (int j = 0; j < 4; ++j) {
        float d = fabsf(lp[j] - t4[j]);
        lloc += (d < 1.f) ? 0.5f * d * d : (d - 0.5f);
      }
    }
  }
  redf[tid] = lloc; redi[tid] = lpos;
  __syncthreads();
  for (int s = TPB / 2; s > 0; s >>= 1) {
    if (tid < s) { redf[tid] += redf[tid + s]; redi[tid] += redi[tid + s]; }
    __syncthreads();
  }
  if (tid == 0) { num_pos[b] = redi[0]; atomicAdd(&accum[0], redf[0]); }
}

// ---------------------------------------------------------------------------
// Kernel 2: g[b,p] = logsumexp(conf[b,p,:]) - conf[b,p,conf_t].
// One wave per 16-prior tile. Sum-of-exp over classes done on the matrix
// pipe: six V_WMMA_F32_16X16X4_F32 with B = ones computes exact f32 row sums.
// A-layout per ISA: lanes 0-15 hold K={0,1}, lanes 16-31 hold K={2,3}.
// ---------------------------------------------------------------------------
__global__ void lse_kernel(const float* __restrict__ conf,
                           const int* __restrict__ conf_t,
                           float* __restrict__ gvals,
                           int P, int C) {
  __shared__ float ssum[8 * 16];
  const int b = blockIdx.y;
  const int wid = threadIdx.x >> 5;
  const int lane = threadIdx.x & 31;
  const int half = lane >> 4;
  const int tile = blockIdx.x * 8 + wid;
  const int rrow = tile * 16 + (lane & 15);
  const int row = (rrow < P) ? rrow : (P - 1);  // clamp; no divergent return
  const float* rowp = conf + ((size_t)b * P + row) * C;

  float xv[12];
  float m = -3.402823466e38f;
#pragma unroll
  for (int k = 0; k < 6; ++k) {
    int c = 4 * k + 2 * half;
    float a0 = (c < C) ? rowp[c] : 0.f;
    float a1 = (c + 1 < C) ? rowp[c + 1] : 0.f;
    xv[2 * k] = a0; xv[2 * k + 1] = a1;
    if (c < C) m = fmaxf(m, a0);
    if (c + 1 < C) m = fmaxf(m, a1);
  }
  // combine row max across the two half-waves handling the same 16 rows
  m = fmaxf(m, __shfl_xor(m, 16, 32));

#if __has_builtin(__builtin_amdgcn_wmma_f32_16x16x4_f32)
  v8f acc = {0.f, 0.f, 0.f, 0.f, 0.f, 0.f, 0.f, 0.f};
  v2f ones; ones.x = 1.f; ones.y = 1.f;
#pragma unroll
  for (int k = 0; k < 6; ++k) {
    int c = 4 * k + 2 * half;
    v2f a;
    a.x = (c < C) ? __expf(xv[2 * k] - m) : 0.f;
    a.y = (c + 1 < C) ? __expf(xv[2 * k + 1] - m) : 0.f;
    // D[m][n] = sum_k A[m][k] * 1 + C : exact f32 row-sum on the matrix pipe
    acc = __builtin_amdgcn_wmma_f32_16x16x4_f32(false, a, false, ones, (short)0,
                                                acc, false, false);
  }
  // D layout: lane<16 VGPR r = row r; lane>=16 VGPR r = row r+8 (all N equal).
  if ((lane & 15) == 0) {
    int base = wid * 16 + half * 8;
#pragma unroll
    for (int r = 0; r < 8; ++r) ssum[base + r] = acc[r];
  }
  __syncthreads();
  float s = ssum[wid * 16 + (lane & 15)];
#else
  float s = 0.f;
#pragma unroll
  for (int k = 0; k < 6; ++k) {
    int c = 4 * k + 2 * half;
    if (c < C) s += __expf(xv[2 * k] - m);
    if (c + 1 < C) s += __expf(xv[2 * k + 1] - m);
  }
  s += __shfl_xor(s, 16, 32);
  __syncthreads();
#endif
  float lse = __logf(s) + m;
  if (half == 0 && rrow < P) {
    int ct = conf_t[(size_t)b * P + rrow];
    gvals[(size_t)b * P + rrow] = lse - rowp[ct];
  }
}

// ---------------------------------------------------------------------------
// Kernel 3: hard-negative mining via exact radix-select on float bits, O(P).
// sum_{rank<k} lv  ==  sum_{lv>t} lv + (k-m)*t   (ties contribute t each;
// index-stable tie order is irrelevant to the SUM; zero-ties contribute 0).
// ---------------------------------------------------------------------------
__global__ void mine_kernel(const int* __restrict__ conf_t,
                            const float* __restrict__ gvals,
                            const int* __restrict__ num_pos,
                            float* __restrict__ accum,
                            int P) {
  extern __shared__ unsigned char smem[];
  float* sv = (float*)smem;
  int* hist = (int*)(smem + (size_t)P * sizeof(float));
  __shared__ unsigned s_prefix;
  __shared__ int s_m;
  __shared__ float redf[TPB];

  const int b = blockIdx.x;
  const int tid = threadIdx.x;
  const int np = num_pos[b];
  int k = 3 * np;
  if (k > P - 1) k = P - 1;

  float psum = 0.f;
  for (int p = tid; p < P; p += TPB) {
    int cf = conf_t[(size_t)b * P + p];
    float g = gvals[(size_t)b * P + p];
    bool pos = cf > 0;
    sv[p] = pos ? 0.f : g;   // g >= 0 always (lse >= any logit)
    if (pos) psum += g;      // positives always selected: add their NLL
  }
  __syncthreads();

  float negsum = 0.f;
  if (k > 0) {  // k uniform per block -> barriers legal
    // level 0: bits[31:21]
    for (int i = tid; i < 2048; i += TPB) hist[i] = 0;
    __syncthreads();
    for (int p = tid; p < P; p += TPB)
      atomicAdd(&hist[__float_as_uint(sv[p]) >> 21], 1);
    __syncthreads();
    if (tid == 0) {
      int m = 0, chosen = 0;
      for (int bin = 2047; bin >= 0; --bin) {
        int c = hist[bin];
        if (m + c >= k) { chosen = bin; break; }
        m += c;
      }
      s_prefix = (unsigned)chosen; s_m = m;
    }
    __syncthreads();
    unsigned pre0 = s_prefix;
    // level 1: bits[20:10]
    for (int i = tid; i < 2048; i += TPB) hist[i] = 0;
    __syncthreads();
    for (int p = tid; p < P; p += TPB) {
      unsigned bt = __float_as_uint(sv[p]);
      if ((bt >> 21) == pre0) atomicAdd(&hist[(bt >> 10) & 0x7FFu], 1);
    }
    __syncthreads();
    if (tid == 0) {
      int m = s_m, chosen = 0;
      for (int bin = 2047; bin >= 0; --bin) {
        int c = hist[bin];
        if (m + c >= k) { chosen = bin; break; }
        m += c;
      }
      s_prefix = (pre0 << 11) | (unsigned)chosen; s_m = m;
    }
    __syncthreads();
    unsigned pre1 = s_prefix;  // 22-bit prefix
    // level 2: bits[9:0]
    for (int i = tid; i < 1024; i += TPB) hist[i] = 0;
    __syncthreads();
    for (int p = tid; p < P; p += TPB) {
      unsigned bt = __float_as_uint(sv[p]);
      if ((bt >> 10) == pre1) atomicAdd(&hist[bt & 0x3FFu], 1);
    }
    __syncthreads();
    if (tid == 0) {
      int m = s_m, chosen = 0;
      for (int bin = 1023; bin >= 0; --bin) {
        int c = hist[bin];
        if (m + c >= k) { chosen = bin; break; }
        m += c;
      }
      s_prefix = (pre1 << 10) | (unsigned)chosen; s_m = m;
    }
    __syncthreads();
    const unsigned tbits = s_prefix;       // exact k-th largest bit pattern
    const int m = s_m;                     // strictly-greater count
    const float t = __uint_as_float(tbits);
    float gsum = 0.f;
    for (int p = tid; p < P; p += TPB) {
      float v = sv[p];
      if (__float_as_uint(v) > tbits) gsum += v;
    }
    redf[tid] = gsum; __syncthreads();
    for (int s = TPB / 2; s > 0; s >>= 1) {
      if (tid < s) redf[tid] += redf[tid + s];
      __syncthreads();
    }
    if (tid == 0) negsum = redf[0] + (float)(k - m) * t;
    __syncthreads();
  }
  redf[tid] = psum; __syncthreads();
  for (int s = TPB / 2; s > 0; s >>= 1) {
    if (tid < s) redf[tid] += redf[tid + s];
    __syncthreads();
  }
  if (tid == 0) atomicAdd(&accum[1], redf[0] + negsum);
}

__global__ void init_kernel(float* accum) {
  if (threadIdx.x == 0) { accum[0] = 0.f; accum[1] = 0.f; }
}

__global__ void finalize_kernel(const int* __restrict__ num_pos,
                                const float* __restrict__ accum,
                                float* __restrict__ out, int B) {
  __shared__ int redi[TPB];
  int tid = threadIdx.x;
  int v = 0;
  for (int i = tid; i < B; i += TPB) v += num_pos[i];
  redi[tid] = v; __syncthreads();
  for (int s = TPB / 2; s > 0; s >>= 1) {
    if (tid < s) redi[tid] += redi[tid + s];
    __syncthreads();
  }
  if (tid == 0) {
    float N = (float)redi[0];
    out[0] = accum[0] / N;   // ALPHA = 1
    out[1] = accum[1] / N;
  }
}

extern "C" void kernel_launch(void* const* d_in, const int* in_sizes, int n_in,
                              void* d_out, int out_size, void* d_ws, size_t ws_size,
                              hipStream_t stream) {
  const float* conf   = (const float*)d_in[0];
  const float* loc    = (const float*)d_in[1];
  const float* priors = (const float*)d_in[2];
  const float* labels = (const float*)d_in[3];
  const int*   objc   = (const int*)d_in[4];

  const int B = in_sizes[4];                                   // obj_count: [B,1]
  const int P = in_sizes[2] / 4;                               // prior_boxes: [P,4]
  const int C = (int)(in_sizes[0] / ((long long)B * (long long)P));
  const int NMAX = in_sizes[3] / (B * 5);                      // labels: [B,NMAX,5]

  size_t BP = (size_t)B * P;
  char* ws = (char*)d_ws;
  int*   conf_t  = (int*)ws;    ws += BP * sizeof(int);
  float* gvals   = (float*)ws;  ws += BP * sizeof(float);
  int*   num_pos = (int*)ws;    ws += (((size_t)B * sizeof(int)) + 255) / 256 * 256;
  float* accum   = (float*)ws;  // [2]

  float* out = (float*)d_out;

  hipLaunchKernelGGL(init_kernel, dim3(1), dim3(32), 0, stream, accum);
  hipLaunchKernelGGL(match_kernel, dim3(B), dim3(TPB), 0, stream,
                     priors, labels, objc, loc, conf_t, num_pos, accum, P, NMAX);
  int tiles = (P + 15) / 16;
  int bx = (tiles + 7) / 8;  // 8 waves (8 tiles) per block
  hipLaunchKernelGGL(lse_kernel, dim3(bx, B), dim3(TPB), 0, stream,
                     conf, conf_t, gvals, P, C);
  size_t sh3 = (size_t)P * sizeof(float) + 2048 * sizeof(int);
  hipLaunchKernelGGL(mine_kernel, dim3(B), dim3(TPB), sh3, stream,
                     conf_t, gvals, num_pos, accum, P);
  hipLaunchKernelGGL(finalize_kernel, dim3(1), dim3(TPB), 0, stream,
                     num_pos, accum, out, B);
}